// APPNP_model_38173669327326
// MI455X (gfx1250) — compile-verified
//
#include <hip/hip_runtime.h>
#include <hip/hip_bf16.h>

typedef __attribute__((ext_vector_type(2))) float v2f;
typedef __attribute__((ext_vector_type(8))) float v8f;

#define ALPHA 0.1f
#define ONE_MINUS_ALPHA 0.9f
#define IN_FEATS 256
#define HID_FEATS 256
#define OUT_FEATS 3

// ---------------------------------------------------------------------------
// Encoder: h0 = relu(x @ W1 + b1) @ W2 + b2
// 512 threads = 16 wave32 per block; each block handles 16 rows of x.
// Each wave computes a 16x16 tile of the hidden layer via 64 chained
// V_WMMA_F32_16X16X4_F32 ops (K = 256), exact fp32.
// ---------------------------------------------------------------------------
__global__ __launch_bounds__(512) void encoder_kernel(
    const float* __restrict__ x, const float* __restrict__ W1,
    const float* __restrict__ b1, const float* __restrict__ W2,
    const float* __restrict__ b2, float* __restrict__ h0, int n_nodes) {
  __shared__ float Xs[16][IN_FEATS + 1];   // +1 pad: conflict-free column reads
  __shared__ float Hs[16][HID_FEATS + 1];

  const int tid = threadIdx.x;
  const int row0 = blockIdx.x * 16;

  // Stage 16x256 x-tile into LDS (clamped reads for the ragged tail).
  for (int i = tid; i < 16 * IN_FEATS; i += 512) {
    int r = i >> 8, c = i & 255;
    int rg = row0 + r;
    if (rg >= n_nodes) rg = n_nodes - 1;
    Xs[r][c] = x[(long long)rg * IN_FEATS + c];
  }
  __syncthreads();

  const int wave = tid >> 5;          // 0..15 -> hidden-column tile
  const int lane = tid & 31;
  const int lmod = lane & 15;         // row (A) / col (B,D) within tile
  const int khalf = (lane >> 4) << 1; // lanes 0-15: K+{0,1}; lanes 16-31: K+{2,3}
  const int colbase = wave * 16;

  v8f acc = {};
  for (int kk = 0; kk < IN_FEATS; kk += 4) {
    v2f a, b;
    // A fragment (16x4, M = lmod): VGPR0 = K+khalf, VGPR1 = K+khalf+1
    a.x = Xs[lmod][kk + khalf];
    a.y = Xs[lmod][kk + khalf + 1];
    // B fragment (4x16, N = colbase+lmod): row K striped across lanes per VGPR
    b.x = W1[(kk + khalf) * HID_FEATS + colbase + lmod];
    b.y = W1[(kk + khalf + 1) * HID_FEATS + colbase + lmod];
    acc = __builtin_amdgcn_wmma_f32_16x16x4_f32(false, a, false, b,
                                                (short)0, acc, false, false);
  }

  // D layout: VGPR v -> M = v (lanes 0-15) or v+8 (lanes 16-31), N = lmod.
  const float bias = b1[colbase + lmod];
  const int mbase = (lane >> 4) << 3;
#pragma unroll
  for (int v = 0; v < 8; ++v) {
    float val = acc[v] + bias;
    Hs[mbase + v][colbase + lmod] = val > 0.f ? val : 0.f;
  }
  __syncthreads();

  // Tiny second GEMM: [16 x 256] @ [256 x 3] from LDS.
  if (tid < 16 * OUT_FEATS) {
    int r = tid / OUT_FEATS, c = tid % OUT_FEATS;
    if (row0 + r < n_nodes) {
      float s = b2[c];
      for (int k = 0; k < HID_FEATS; ++k) s += Hs[r][k] * W2[k * OUT_FEATS + c];
      h0[(long long)(row0 + r) * OUT_FEATS + c] = s;
    }
  }
}

// ---------------------------------------------------------------------------
// gcn_norm precompute
// ---------------------------------------------------------------------------
__global__ void deg_init_kernel(float* __restrict__ deg, int n) {
  int i = blockIdx.x * 256 + threadIdx.x;
  if (i < n) deg[i] = 1.0f;  // self loop
}

__global__ void deg_count_kernel(const long long* __restrict__ dst,
                                 float* __restrict__ deg, long long e_count) {
  long long e = (long long)blockIdx.x * 256 + threadIdx.x;
  if (e < e_count) atomicAdd(&deg[(int)dst[e]], 1.0f);
}

__global__ void dinv_kernel(float* __restrict__ deg_to_dinv,
                            float* __restrict__ selfw, int n) {
  int i = blockIdx.x * 256 + threadIdx.x;
  if (i >= n) return;
  float d = deg_to_dinv[i];            // >= 1 (self loop)
  float di = __builtin_amdgcn_rsqf(d);
  deg_to_dinv[i] = di;                 // deg buffer becomes dinv in place
  selfw[i] = ONE_MINUS_ALPHA * di * di;
}

__global__ void norm_kernel(const long long* __restrict__ src,
                            const long long* __restrict__ dst,
                            const float* __restrict__ dinv,
                            float* __restrict__ normw, long long e_count) {
  long long e = (long long)blockIdx.x * 256 + threadIdx.x;
  if (e < e_count)
    normw[e] = ONE_MINUS_ALPHA * dinv[(int)src[e]] * dinv[(int)dst[e]];
}

// ---------------------------------------------------------------------------
// APPNP step: nxt = alpha*h0 + selfw*cur  (per node), then edge scatter-add.
// All hot data (edge_index 51MB, normw 13MB, h buffers 1.2MB) lives in the
// 192MB L2 -> iterations are L2-bandwidth bound, negligible HBM traffic.
// ---------------------------------------------------------------------------
__global__ void appnp_init_kernel(const float* __restrict__ h0,
                                  const float* __restrict__ cur,
                                  const float* __restrict__ selfw,
                                  float* __restrict__ nxt, int n) {
  int i = blockIdx.x * 256 + threadIdx.x;
  if (i >= n) return;
  float sw = selfw[i];
#pragma unroll
  for (int c = 0; c < OUT_FEATS; ++c)
    nxt[i * OUT_FEATS + c] = ALPHA * h0[i * OUT_FEATS + c] + sw * cur[i * OUT_FEATS + c];
}

__global__ void edge_prop_kernel(const long long* __restrict__ src,
                                 const long long* __restrict__ dst,
                                 const float* __restrict__ normw,
                                 const float* __restrict__ cur,
                                 float* __restrict__ nxt, long long e_count) {
  long long e = (long long)blockIdx.x * 256 + threadIdx.x;
  if (e >= e_count) return;
  int s = (int)src[e];
  int d = (int)dst[e];
  float w = normw[e];
  float m0 = w * cur[s * OUT_FEATS + 0];
  float m1 = w * cur[s * OUT_FEATS + 1];
  float m2 = w * cur[s * OUT_FEATS + 2];
  atomicAdd(&nxt[d * OUT_FEATS + 0], m0);
  atomicAdd(&nxt[d * OUT_FEATS + 1], m1);
  atomicAdd(&nxt[d * OUT_FEATS + 2], m2);
}

// ---------------------------------------------------------------------------
extern "C" void kernel_launch(void* const* d_in, const int* in_sizes, int n_in,
                              void* d_out, int out_size, void* d_ws, size_t ws_size,
                              hipStream_t stream) {
  const float* x = (const float*)d_in[0];
  const long long* ei = (const long long*)d_in[1];
  const float* W1 = (const float*)d_in[2];
  const float* b1 = (const float*)d_in[3];
  const float* W2 = (const float*)d_in[4];
  const float* b2 = (const float*)d_in[5];
  float* out = (float*)d_out;

  const int N = in_sizes[0] / IN_FEATS;
  const long long E = (long long)in_sizes[1] / 2;
  const long long* srcv = ei;
  const long long* dstv = ei + E;

  // Workspace layout (floats): H0[3N] | A[3N] | dinv[N] | selfw[N] | normw[E]
  float* ws = (float*)d_ws;
  float* H0    = ws;
  float* A     = ws + 3LL * N;
  float* dinv  = ws + 6LL * N;   // holds deg first, then dinv in place
  float* selfw = ws + 7LL * N;
  float* normw = ws + 8LL * N;

  const unsigned nbN = (unsigned)((N + 255) / 256);
  const unsigned nbE = (unsigned)((E + 255) / 256);
  const unsigned nbEnc = (unsigned)((N + 15) / 16);

  // Encoder (WMMA fp32) -> H0
  encoder_kernel<<<nbEnc, 512, 0, stream>>>(x, W1, b1, W2, b2, H0, N);

  // gcn_norm precompute
  deg_init_kernel<<<nbN, 256, 0, stream>>>(dinv, N);
  deg_count_kernel<<<nbE, 256, 0, stream>>>(dstv, dinv, E);
  dinv_kernel<<<nbN, 256, 0, stream>>>(dinv, selfw, N);
  norm_kernel<<<nbE, 256, 0, stream>>>(srcv, dstv, dinv, normw, E);

  // K = 10 propagation steps; ping-pong A <-> out so step 10 lands in d_out.
  const float* cur = H0;
  float* nxt = A;
  for (int t = 0; t < 10; ++t) {
    appnp_init_kernel<<<nbN, 256, 0, stream>>>(H0, cur, selfw, nxt, N);
    edge_prop_kernel<<<nbE, 256, 0, stream>>>(srcv, dstv, normw, cur, nxt, E);
    cur = nxt;
    nxt = (t % 2 == 0) ? out : A;
  }
}